// Top2Gate_6124623364426
// MI455X (gfx1250) — compile-verified
//
#include <hip/hip_runtime.h>
#include <hip/hip_bf16.h>

// Top-2 MoE gate, fused: logits = x @ W^T (+b), softmax, top-2, renorm, entropy.
// N=65536 tokens, D=2048, K=64 experts. fp32 end-to-end via V_WMMA_F32_16X16X4_F32.
// HBM-bound on the 512MB x read (~23us @ 23.3 TB/s). Each wave owns 32 tokens
// (two A tiles) x all 64 experts (4 B tiles): 6 b64 loads feed 8 WMMAs per
// K-step, W fragments stay L0-resident (shared by all 8 waves), x streams.

#define NTOK 65536
#define DDIM 2048
#define KEXP 64
#define ROWP 66   // LDS row pad: bank = (2*lane + j) % 64 -> conflict-free for 32 lanes

typedef float v2f __attribute__((ext_vector_type(2)));
typedef float v8f __attribute__((ext_vector_type(8)));

__global__ __launch_bounds__(256) void top2gate_kernel(
    const float* __restrict__ x, const float* __restrict__ W,
    const float* __restrict__ bias, float* __restrict__ out)
{
    // 8 waves/block, 32 tokens/wave -> 256 tokens/block.
    __shared__ float smem[8][32][ROWP];   // 67.6 KB

    const int lane = threadIdx.x & 31;
    const int wave = threadIdx.x >> 5;
    const int m    = lane & 15;          // row inside 16x16 tile
    const int kk   = (lane >> 4) * 2;    // A/B K-offset: lanes 0-15 -> K=0,1 ; 16-31 -> K=2,3
    const int tokenBase = blockIdx.x * 256 + wave * 32;

    const float* xrow0 = x + (size_t)(tokenBase +  0 + m) * DDIM + kk;
    const float* xrow1 = x + (size_t)(tokenBase + 16 + m) * DDIM + kk;
    const float* wrow0 = W + (size_t)( 0 + m) * DDIM + kk;
    const float* wrow1 = W + (size_t)(16 + m) * DDIM + kk;
    const float* wrow2 = W + (size_t)(32 + m) * DDIM + kk;
    const float* wrow3 = W + (size_t)(48 + m) * DDIM + kk;

    v8f accA0 = {}, accA1 = {}, accA2 = {}, accA3 = {};   // tokens 0-15  x experts 0..63
    v8f accB0 = {}, accB1 = {}, accB2 = {}, accB3 = {};   // tokens 16-31 x experts 0..63

    #pragma unroll 2
    for (int d0 = 0; d0 < DDIM; d0 += 4) {
        v2f a0 = *(const v2f*)(xrow0 + d0);
        v2f a1 = *(const v2f*)(xrow1 + d0);
        v2f b0 = *(const v2f*)(wrow0 + d0);
        v2f b1 = *(const v2f*)(wrow1 + d0);
        v2f b2 = *(const v2f*)(wrow2 + d0);
        v2f b3 = *(const v2f*)(wrow3 + d0);
        accA0 = __builtin_amdgcn_wmma_f32_16x16x4_f32(false, a0, false, b0, (short)0, accA0, false, false);
        accA1 = __builtin_amdgcn_wmma_f32_16x16x4_f32(false, a0, false, b1, (short)0, accA1, false, false);
        accA2 = __builtin_amdgcn_wmma_f32_16x16x4_f32(false, a0, false, b2, (short)0, accA2, false, false);
        accA3 = __builtin_amdgcn_wmma_f32_16x16x4_f32(false, a0, false, b3, (short)0, accA3, false, false);
        accB0 = __builtin_amdgcn_wmma_f32_16x16x4_f32(false, a1, false, b0, (short)0, accB0, false, false);
        accB1 = __builtin_amdgcn_wmma_f32_16x16x4_f32(false, a1, false, b1, (short)0, accB1, false, false);
        accB2 = __builtin_amdgcn_wmma_f32_16x16x4_f32(false, a1, false, b2, (short)0, accB2, false, false);
        accB3 = __builtin_amdgcn_wmma_f32_16x16x4_f32(false, a1, false, b3, (short)0, accB3, false, false);
    }

    // C/D layout: VGPR r -> lanes 0-15: token=r, expert=lane; lanes 16-31: token=r+8, expert=lane-16.
    const int trow = (lane >> 4) * 8;
    const int n    = lane & 15;
    #pragma unroll
    for (int r = 0; r < 8; ++r) {
        smem[wave][trow + r][ 0 + n] = accA0[r];
        smem[wave][trow + r][16 + n] = accA1[r];
        smem[wave][trow + r][32 + n] = accA2[r];
        smem[wave][trow + r][48 + n] = accA3[r];
        smem[wave][16 + trow + r][ 0 + n] = accB0[r];
        smem[wave][16 + trow + r][16 + n] = accB1[r];
        smem[wave][16 + trow + r][32 + n] = accB2[r];
        smem[wave][16 + trow + r][48 + n] = accB3[r];
    }
    __syncthreads();

    // Epilogue: every lane owns one token (full 64-expert row).
    {
        const int token = tokenBase + lane;
        const float* row = &smem[wave][lane][0];

        float lg[KEXP];
        float v1 = -3.4e38f, v2 = -3.4e38f;
        int   i1 = 0,        i2 = 0;
        #pragma unroll
        for (int j = 0; j < KEXP; ++j) {
            float t = row[j] + bias[j];   // TEMPERATURE == 1.0
            lg[j] = t;
            if (t > v1)      { v2 = v1; i2 = i1; v1 = t; i1 = j; }
            else if (t > v2) { v2 = t; i2 = j; }
        }

        float S = 0.0f;
        #pragma unroll
        for (int j = 0; j < KEXP; ++j) S += __expf(lg[j] - v1);

        const float invS  = 1.0f / S;
        const float p1    = invS;                      // exp(0)/S
        const float p2    = __expf(v2 - v1) * invS;
        const float denom = p1 + p2 + 1e-9f;
        const float q1    = p1 / denom;
        const float q2    = p2 / denom;
        const float ent   = -(q1 * __logf(fmaxf(q1, 1e-12f)) +
                              q2 * __logf(fmaxf(q2, 1e-12f)));

        float* top2p = out;                                  // [N*K]
        float* entp  = out + (size_t)NTOK * KEXP;            // [N]
        float* lgp   = entp + NTOK;                          // [N*K]
        float* t2row = top2p + (size_t)token * KEXP;
        float* lrow  = lgp   + (size_t)token * KEXP;

        #pragma unroll
        for (int j = 0; j < KEXP; ++j) {
            lrow[j]  = lg[j];
            t2row[j] = (j == i1) ? q1 : ((j == i2) ? q2 : 0.0f);
        }
        entp[token] = ent;
    }
}

extern "C" void kernel_launch(void* const* d_in, const int* in_sizes, int n_in,
                              void* d_out, int out_size, void* d_ws, size_t ws_size,
                              hipStream_t stream) {
    const float* x = (const float*)d_in[0];   // [N, D]
    const float* W = (const float*)d_in[1];   // [K, D]
    const float* b = (const float*)d_in[2];   // [K]
    float* out = (float*)d_out;               // top2_probs | entropy | logits

    dim3 grid(NTOK / 256);    // 256 blocks x 256 threads (8 waves, 32 tokens each)
    dim3 block(256);
    top2gate_kernel<<<grid, block, 0, stream>>>(x, W, b, out);
}